// SketchRenderNet_60241211294420
// MI455X (gfx1250) — compile-verified
//
#include <hip/hip_runtime.h>
#include <math.h>

#define IMSIZE 1024
#define NSEG   128
#define FEPS   1e-10f

typedef __attribute__((ext_vector_type(2))) float v2f;
typedef __attribute__((ext_vector_type(8))) float v8f;

// ---------------------------------------------------------------------------
// Render kernel: each block = 256 threads = 8 waves (wave32).
//   blockIdx.x selects a 16-pixel-wide column strip, blockIdx.y a 64-row band.
//   Each wave handles rows (ybase + wave + 8*r), r = 0..7.
//   Per row: 8 x V_WMMA_F32_16X16X4_F32 compute x_gen for 16 segs x 16 px,
//   VALU does the med3-clamp / sqrt tail, shfl_xor(16) folds segment halves.
// ---------------------------------------------------------------------------
__global__ __launch_bounds__(256) void sketch_render_kernel(
    const float* __restrict__ xs, const float* __restrict__ ys,
    float* __restrict__ out, float* __restrict__ bmin, float* __restrict__ bmax)
{
    __shared__ float s_cy[NSEG], s_cx[NSEG], s_c0[NSEG];
    __shared__ float s_sl[NSEG], s_in[NSEG], s_vt[NSEG];
    __shared__ float s_xl[NSEG], s_xh[NSEG], s_yl[NSEG], s_yh[NSEG];
    __shared__ float s_red[16];

    const int t = threadIdx.x;

    // --- per-segment coefficients (threads 0..127) ---
    if (t < NSEG) {
        float x0 = xs[t], x1 = xs[t + 1];
        float y0 = ys[t], y1 = ys[t + 1];
        float dx = x1 - x0, dy = y1 - y0;
        float dx2 = dx * dx;
        float d = dx2 + dy * dy;
        float invd = 1.0f / (d + FEPS);
        float cross = y0 * x1 - x0 * y1;      // y0*x1 - x0*y1
        s_cy[t] = dx * dy * invd;             // coeff of yp in x_gen
        s_cx[t] = dx2 * invd;                 // coeff of xp in x_gen
        s_c0[t] = -dy * cross * invd;         // constant term
        float invdx = 1.0f / (dx + FEPS);
        s_sl[t] = dy * invdx;                 // slope for y_gen
        s_in[t] = cross * invdx;              // intercept for y_gen
        s_vt[t] = (x0 == x1) ? 1.0f : 0.0f;   // vertical flag
        s_xl[t] = fminf(x0, x1); s_xh[t] = fmaxf(x0, x1);
        s_yl[t] = fminf(y0, y1); s_yh[t] = fmaxf(y0, y1);
    }
    __syncthreads();

    const int lane  = t & 31;
    const int wave  = t >> 5;        // 0..7
    const int laneN = lane & 15;     // pixel column / segment M within tile
    const int hi    = lane >> 4;     // 0 = lanes 0-15 (K=0,1), 1 = lanes 16-31 (K=2,3)
    const int xbase = blockIdx.x * 16;
    const int ybase = blockIdx.y * 64;
    const float xp  = (float)(xbase + laneN);

    // A fragments for the 8 segment groups:
    //   lanes 0-15 : VGPR0 = K0 = c_y[m], VGPR1 = K1 = c_x[m]
    //   lanes 16-31: VGPR0 = K2 = c_0[m], VGPR1 = K3 = 0
    v2f afrag[8];
#pragma unroll
    for (int g = 0; g < 8; ++g) {
        int m = g * 16 + laneN;
        afrag[g].x = hi ? s_c0[m] : s_cy[m];
        afrag[g].y = hi ? 0.0f    : s_cx[m];
    }

    float lmin = INFINITY, lmax = -INFINITY;

#pragma unroll 1
    for (int r = 0; r < 8; ++r) {
        const int   row = ybase + wave + r * 8;
        const float yp  = (float)row;

        // B fragment: column N = [yp, xp, 1, 0]
        v2f bfrag;
        bfrag.x = hi ? 1.0f : yp;
        bfrag.y = hi ? 0.0f : xp;

        float acc = 0.0f;
#pragma unroll 2
        for (int g = 0; g < 8; ++g) {
            v8f c = {0.f, 0.f, 0.f, 0.f, 0.f, 0.f, 0.f, 0.f};
            // D(16x16) = A(16x4) x B(4x16) : x_gen for 16 segs x 16 pixels
            v8f dmat = __builtin_amdgcn_wmma_f32_16x16x4_f32(
                false, afrag[g], false, bfrag, (short)0, c, false, false);
            const int mbase = g * 16 + hi * 8;
#pragma unroll
            for (int j = 0; j < 8; ++j) {
                const int m = mbase + j;           // segment index for this row of D
                float xg   = dmat[j];
                float ygen = fmaf(s_sl[m], xg, s_in[m]);
                float ysel = (s_vt[m] != 0.0f) ? yp : ygen;
                // clip == med3 since lo <= hi always
                float xc   = __builtin_amdgcn_fmed3f(xg,   s_xl[m], s_xh[m]);
                float yc   = __builtin_amdgcn_fmed3f(ysel, s_yl[m], s_yh[m]);
                float ddx  = xp - xc;
                float ddy  = yp - yc;
                acc += __builtin_amdgcn_sqrtf(fmaf(ddx, ddx, fmaf(ddy, ddy, FEPS)));
            }
        }
        // lanes L and L^16 hold the same pixel, disjoint segment halves
        acc += __shfl_xor(acc, 16, 32);          // all 32 lanes end up with full sum
        lmin = fminf(lmin, acc);
        lmax = fmaxf(lmax, acc);
        if (lane < 16)
            out[row * IMSIZE + xbase + laneN] = acc;
    }

    // wave min/max reduce (duplicated halves don't affect min/max)
#pragma unroll
    for (int off = 16; off >= 1; off >>= 1) {
        lmin = fminf(lmin, __shfl_xor(lmin, off, 32));
        lmax = fmaxf(lmax, __shfl_xor(lmax, off, 32));
    }
    if (lane == 0) { s_red[wave] = lmin; s_red[8 + wave] = lmax; }
    __syncthreads();
    if (t == 0) {
        float m0 = s_red[0], m1 = s_red[8];
        for (int w = 1; w < 8; ++w) {
            m0 = fminf(m0, s_red[w]);
            m1 = fmaxf(m1, s_red[8 + w]);
        }
        const int bid = blockIdx.y * gridDim.x + blockIdx.x;
        bmin[bid] = m0;
        bmax[bid] = m1;
    }
}

// ---------------------------------------------------------------------------
// Single-block reduction of per-block min/max -> mm[0]=tmin, mm[1]=tmax
// ---------------------------------------------------------------------------
__global__ __launch_bounds__(256) void sketch_minmax_kernel(
    const float* __restrict__ bmin, const float* __restrict__ bmax,
    float* __restrict__ mm, int n)
{
    __shared__ float smin[8], smax[8];
    const int t = threadIdx.x;
    float m0 = INFINITY, m1 = -INFINITY;
    for (int i = t; i < n; i += 256) {
        m0 = fminf(m0, bmin[i]);
        m1 = fmaxf(m1, bmax[i]);
    }
#pragma unroll
    for (int off = 16; off >= 1; off >>= 1) {
        m0 = fminf(m0, __shfl_xor(m0, off, 32));
        m1 = fmaxf(m1, __shfl_xor(m1, off, 32));
    }
    if ((t & 31) == 0) { smin[t >> 5] = m0; smax[t >> 5] = m1; }
    __syncthreads();
    if (t == 0) {
        for (int w = 1; w < 8; ++w) {
            m0 = fminf(m0, smin[w]);
            m1 = fmaxf(m1, smax[w]);
        }
        mm[0] = m0;
        mm[1] = m1;
    }
}

// ---------------------------------------------------------------------------
// In-place global normalization: out = (out - tmin) / (tmax - tmin)
// ---------------------------------------------------------------------------
__global__ void sketch_normalize_kernel(float* __restrict__ out,
                                        const float* __restrict__ mm, int n)
{
    const int i = blockIdx.x * blockDim.x + threadIdx.x;
    if (i < n) {
        const float tmin = mm[0];
        const float tmax = mm[1];
        out[i] = (out[i] - tmin) / (tmax - tmin);
    }
}

extern "C" void kernel_launch(void* const* d_in, const int* in_sizes, int n_in,
                              void* d_out, int out_size, void* d_ws, size_t ws_size,
                              hipStream_t stream)
{
    const float* xs = (const float*)d_in[0];
    const float* ys = (const float*)d_in[1];
    float* out = (float*)d_out;
    float* ws  = (float*)d_ws;

    float* bmin = ws;            // 1024 floats
    float* bmax = ws + 1024;     // 1024 floats
    float* mm   = ws + 2048;     // 2 floats

    dim3 grid(IMSIZE / 16, IMSIZE / 64);   // 64 x 16 = 1024 blocks
    sketch_render_kernel<<<grid, 256, 0, stream>>>(xs, ys, out, bmin, bmax);

    const int nblocks = (IMSIZE / 16) * (IMSIZE / 64);
    sketch_minmax_kernel<<<1, 256, 0, stream>>>(bmin, bmax, mm, nblocks);

    const int n = IMSIZE * IMSIZE;
    sketch_normalize_kernel<<<(n + 255) / 256, 256, 0, stream>>>(out, mm, n);
}